// SparseMoEALU_62380105007633
// MI455X (gfx1250) — compile-verified
//
#include <hip/hip_runtime.h>

#define B_N 16384
#define P_N 8
#define D_N 64
#define H_N 128
#define NOPS 37
#define OP_START 16
#define DIV_OP 3
#define MOD_OP 4
#define NITER 16
#define NCASC 7
#define WPB 2                 // waves per block in main kernel

// LDS pitches (dwords) chosen so A-layout reads (row = l16, fixed col) hit distinct banks:
// bank = (l16*PITCH + k) % 64 ; PITCH % 64 == 4 -> lanes spread 4 banks apart (16 distinct).
#define XPITCH 68             // 16x64 tiles (X, Y, Q, K, V)
#define HPITCH 132            // 16x128 tile (H)
#define SPITCH 20             // 16x16 tile  (S)
// per-wave arena (floats): X 1088 | Y 1088 | BIG 2176 (H 2112 / Q 1088 + K 1088) | V 1088 | S 320
#define ARENA 5760

typedef __attribute__((ext_vector_type(2))) float v2f;
typedef __attribute__((ext_vector_type(8))) float v8f;

#define V8F_ZERO {0.f,0.f,0.f,0.f,0.f,0.f,0.f,0.f}

__device__ __forceinline__ v8f wmma4(v2f a, v2f b, v8f c) {
  // D = A(16x4,f32) * B(4x16,f32) + C(16x16,f32)
  return __builtin_amdgcn_wmma_f32_16x16x4_f32(false, a, false, b, (short)0, c, false, false);
}

__device__ __forceinline__ bool is_carry(int o) {
  return (o <= 2) || (o >= 10 && o <= 13);   // {0,1,2,10,11,12,13}
}

// ---------------- FFN: Xout[masked half-rows] = Xin + relu(Xin@W1 + b1)@W2 + b2 ------------
// Xin/Hs/Xout are LDS (pitches XPITCH/HPITCH); W1:64x128, W2:128x64 row-major in global.
// mask bit0 -> rows 0..7 (element 0), bit1 -> rows 8..15 (element 1).
// Each tile uses two independent WMMA accumulator chains (even/odd k) for XDL ILP.
__device__ __forceinline__ void ffn_blk(const float* __restrict__ Xin, float* __restrict__ Hs,
                                        float* __restrict__ Xout,
                                        const float* __restrict__ W1, const float* __restrict__ B1,
                                        const float* __restrict__ W2, const float* __restrict__ B2,
                                        int mask)
{
  const int lane = threadIdx.x & 31;
  const int half = lane >> 4;
  const int l16  = lane & 15;
  __builtin_prefetch(W1, 0, 0);
  __builtin_prefetch(W2, 0, 0);
  // Hoisted A fragments of X (16x64): reused across all 8 N-tiles of GEMM1.
  v2f aX[16];
#pragma unroll
  for (int kk = 0; kk < 16; ++kk)
    aX[kk] = *(const v2f*)(Xin + l16*XPITCH + kk*4 + 2*half);
  // GEMM1: H = relu(X@W1 + b1)      (16x64)@(64x128)
  for (int nt = 0; nt < 8; ++nt) {
    v8f acc0 = V8F_ZERO, acc1 = V8F_ZERO;
#pragma unroll
    for (int kk = 0; kk < 16; kk += 2) {
      v2f b0, b1v;
      b0[0]  = W1[(kk*4 + 2*half        )*128 + nt*16 + l16];
      b0[1]  = W1[(kk*4 + 2*half + 1    )*128 + nt*16 + l16];
      b1v[0] = W1[(kk*4 + 2*half + 4    )*128 + nt*16 + l16];
      b1v[1] = W1[(kk*4 + 2*half + 5    )*128 + nt*16 + l16];
      acc0 = wmma4(aX[kk],     b0,  acc0);
      acc1 = wmma4(aX[kk + 1], b1v, acc1);
    }
    const float bias = B1[nt*16 + l16];
#pragma unroll
    for (int r = 0; r < 8; ++r)
      Hs[(r + 8*half)*HPITCH + nt*16 + l16] = fmaxf(acc0[r] + acc1[r] + bias, 0.f);
  }
  // Hoisted A fragments of H (16x128): reused across all 4 N-tiles of GEMM2.
  v2f aH[32];
#pragma unroll
  for (int kk = 0; kk < 32; ++kk)
    aH[kk] = *(const v2f*)(Hs + l16*HPITCH + kk*4 + 2*half);
  // GEMM2: Xout = Xin + H@W2 + b2    (16x128)@(128x64)
  for (int nt = 0; nt < 4; ++nt) {
    v8f acc0 = V8F_ZERO, acc1 = V8F_ZERO;
#pragma unroll
    for (int kk = 0; kk < 32; kk += 2) {
      v2f b0, b1v;
      b0[0]  = W2[(kk*4 + 2*half    )*64 + nt*16 + l16];
      b0[1]  = W2[(kk*4 + 2*half + 1)*64 + nt*16 + l16];
      b1v[0] = W2[(kk*4 + 2*half + 4)*64 + nt*16 + l16];
      b1v[1] = W2[(kk*4 + 2*half + 5)*64 + nt*16 + l16];
      acc0 = wmma4(aH[kk],     b0,  acc0);
      acc1 = wmma4(aH[kk + 1], b1v, acc1);
    }
    const float bias = B2[nt*16 + l16];
    if (mask & (half ? 2 : 1)) {
#pragma unroll
      for (int r = 0; r < 8; ++r) {
        const int row = r + 8*half;
        Xout[row*XPITCH + nt*16 + l16] =
            Xin[row*XPITCH + nt*16 + l16] + acc0[r] + acc1[r] + bias;
      }
    }
  }
}

// Block-diagonal attention over the pair: Ys = Xs + softmax(QK^T/8, per 8x8 block) @ V
__device__ __forceinline__ void attn_blk(const float* __restrict__ Xs,
                                         float* __restrict__ Qs, float* __restrict__ Ks,
                                         float* __restrict__ Vs, float* __restrict__ Ss,
                                         float* __restrict__ Ys,
                                         const float* __restrict__ Wq, const float* __restrict__ Wk,
                                         const float* __restrict__ Wv)
{
  const int lane = threadIdx.x & 31;
  const int half = lane >> 4;
  const int l16  = lane & 15;
  // Hoisted A fragments of X: shared by the Q, K and V projections.
  v2f aX[16];
#pragma unroll
  for (int kk = 0; kk < 16; ++kk)
    aX[kk] = *(const v2f*)(Xs + l16*XPITCH + kk*4 + 2*half);
  for (int m = 0; m < 3; ++m) {
    const float* W = (m == 0) ? Wq : ((m == 1) ? Wk : Wv);
    float*       O = (m == 0) ? Qs : ((m == 1) ? Ks : Vs);
    for (int nt = 0; nt < 4; ++nt) {
      v8f acc0 = V8F_ZERO, acc1 = V8F_ZERO;
#pragma unroll
      for (int kk = 0; kk < 16; kk += 2) {
        v2f b0, b1v;
        b0[0]  = W[(kk*4 + 2*half    )*64 + nt*16 + l16];
        b0[1]  = W[(kk*4 + 2*half + 1)*64 + nt*16 + l16];
        b1v[0] = W[(kk*4 + 2*half + 4)*64 + nt*16 + l16];
        b1v[1] = W[(kk*4 + 2*half + 5)*64 + nt*16 + l16];
        acc0 = wmma4(aX[kk],     b0,  acc0);
        acc1 = wmma4(aX[kk + 1], b1v, acc1);
      }
#pragma unroll
      for (int r = 0; r < 8; ++r)
        O[(r + 8*half)*XPITCH + nt*16 + l16] = acc0[r] + acc1[r];
    }
  }
  // S = (Q @ K^T) * 1/sqrt(64); B-fragment of K^T: B[k][n] = K[n][k] (same A-style indexing)
  {
    v8f acc0 = V8F_ZERO, acc1 = V8F_ZERO;
#pragma unroll
    for (int kk = 0; kk < 16; kk += 2) {
      v2f a0 = *(const v2f*)(Qs + l16*XPITCH + kk*4 + 2*half);
      v2f b0 = *(const v2f*)(Ks + l16*XPITCH + kk*4 + 2*half);
      v2f a1 = *(const v2f*)(Qs + l16*XPITCH + kk*4 + 4 + 2*half);
      v2f b1v = *(const v2f*)(Ks + l16*XPITCH + kk*4 + 4 + 2*half);
      acc0 = wmma4(a0, b0,  acc0);
      acc1 = wmma4(a1, b1v, acc1);
    }
#pragma unroll
    for (int r = 0; r < 8; ++r)
      Ss[(r + 8*half)*SPITCH + l16] = (acc0[r] + acc1[r]) * 0.125f;
  }
  // Per-row softmax over the valid 8-wide diagonal block; zero the off-diagonal block.
  if (lane < 16) {
    const int row = lane;
    const int c0 = (row < 8) ? 0 : 8;
    const int cz = 8 - c0;
    float m = -3.0e38f;
    float e[8];
#pragma unroll
    for (int j = 0; j < 8; ++j) m = fmaxf(m, Ss[row*SPITCH + c0 + j]);
    float s = 0.f;
#pragma unroll
    for (int j = 0; j < 8; ++j) { e[j] = __expf(Ss[row*SPITCH + c0 + j] - m); s += e[j]; }
    const float inv = 1.0f / s;
#pragma unroll
    for (int j = 0; j < 8; ++j) Ss[row*SPITCH + c0 + j] = e[j] * inv;
#pragma unroll
    for (int j = 0; j < 8; ++j) Ss[row*SPITCH + cz + j] = 0.f;
  }
  // Ys = Xs + S @ V   (16x16)@(16x64); off-diagonal zeros make it exact per element
  v2f aS[4];
#pragma unroll
  for (int kk = 0; kk < 4; ++kk)
    aS[kk] = *(const v2f*)(Ss + l16*SPITCH + kk*4 + 2*half);
  for (int nt = 0; nt < 4; ++nt) {
    v8f acc0 = V8F_ZERO, acc1 = V8F_ZERO;
#pragma unroll
    for (int kk = 0; kk < 4; kk += 2) {
      v2f b0, b1v;
      b0[0]  = Vs[(kk*4 + 2*half    )*XPITCH + nt*16 + l16];
      b0[1]  = Vs[(kk*4 + 2*half + 1)*XPITCH + nt*16 + l16];
      b1v[0] = Vs[(kk*4 + 2*half + 4)*XPITCH + nt*16 + l16];
      b1v[1] = Vs[(kk*4 + 2*half + 5)*XPITCH + nt*16 + l16];
      acc0 = wmma4(aS[kk],     b0,  acc0);
      acc1 = wmma4(aS[kk + 1], b1v, acc1);
    }
#pragma unroll
    for (int r = 0; r < 8; ++r) {
      const int row = r + 8*half;
      Ys[row*XPITCH + nt*16 + l16] =
          Xs[row*XPITCH + nt*16 + l16] + acc0[r] + acc1[r];
    }
  }
}

// ---------------- routing pre-pass ----------------
__global__ void k_zero(int* counts, int* cursor) {
  const int t = threadIdx.x;
  if (t < NOPS) { counts[t] = 0; cursor[t] = 0; }
}

__global__ void k_route(const float* __restrict__ x, int* __restrict__ op, int* __restrict__ counts) {
  const int b = blockIdx.x * blockDim.x + threadIdx.x;
  if (b >= B_N) return;
  const float* r = x + (size_t)b * (P_N * D_N) + OP_START;
  int best = 0; float bv = r[0];
#pragma unroll
  for (int i = 1; i < NOPS; ++i) { const float v = r[i]; if (v > bv) { bv = v; best = i; } }
  op[b] = best;
  atomicAdd(&counts[best], 1);
}

__global__ void k_scan(const int* __restrict__ counts, int* __restrict__ offs) {
  if (threadIdx.x == 0) {
    int a = 0;
    for (int i = 0; i < NOPS; ++i) { offs[i] = a; a += counts[i]; }
    offs[NOPS] = a;
  }
}

__global__ void k_scatter(const int* __restrict__ op, const int* __restrict__ offs,
                          int* __restrict__ cursor, int* __restrict__ perm) {
  const int b = blockIdx.x * blockDim.x + threadIdx.x;
  if (b >= B_N) return;
  const int o = op[b];
  const int pos = offs[o] + atomicAdd(&cursor[o], 1);
  perm[pos] = b;
}

// ---------------- fused main kernel: one wave per same-op pair ----------------
__global__ __launch_bounds__(WPB * 32)
void k_main(const float* __restrict__ x,
            const float* __restrict__ Wi1, const float* __restrict__ bi1,
            const float* __restrict__ Wi2, const float* __restrict__ bi2,
            const float* __restrict__ Wq,  const float* __restrict__ Wk, const float* __restrict__ Wv,
            const float* __restrict__ Wc1, const float* __restrict__ bc1,
            const float* __restrict__ Wc2, const float* __restrict__ bc2,
            const float* __restrict__ Wd1, const float* __restrict__ bd1,
            const float* __restrict__ Wd2, const float* __restrict__ bd2,
            const float* __restrict__ Wm1, const float* __restrict__ bm1,
            const float* __restrict__ Wm2, const float* __restrict__ bm2,
            const float* __restrict__ Wf1, const float* __restrict__ bf1,
            const float* __restrict__ Wf2, const float* __restrict__ bf2,
            float* __restrict__ out, const int* __restrict__ opArr, const int* __restrict__ perm)
{
  __shared__ float smem[WPB * ARENA];
  const int wv   = threadIdx.x >> 5;
  const int lane = threadIdx.x & 31;
  float* base = smem + wv * ARENA;
  float* Xs  = base;               // 16 x 64  (pitch 68)
  float* Ys  = base + 1088;        // 16 x 64
  float* BIG = base + 2176;        // H (16x128, pitch 132) or Q|K
  float* Hs  = BIG;
  float* Qs  = BIG;
  float* Ks  = BIG + 1088;
  float* Vs  = base + 4352;        // 16 x 64
  float* Ss  = base + 5440;        // 16 x 16 (pitch 20)

  const int pair = blockIdx.x * WPB + wv;
  const int e0 = perm[2 * pair + 0];
  const int e1 = perm[2 * pair + 1];
  const int op0 = opArr[e0];
  const int op1 = opArr[e1];
  const int two = (op0 != op1) ? 1 : 0;   // rare boundary pair with mixed ops

  // load pair: rows 0..7 <- x[e0], rows 8..15 <- x[e1]
  {
    const float* s0 = x + (size_t)e0 * 512;
    const float* s1 = x + (size_t)e1 * 512;
    for (int idx = lane; idx < 256; idx += 32) {
      const int row = idx >> 4;
      const int c4  = (idx & 15) * 4;
      const float* src = (row < 8) ? (s0 + row * 64 + c4) : (s1 + (row - 8) * 64 + c4);
      *(float4*)(Xs + row * XPITCH + c4) = *(const float4*)src;
    }
  }

  // Stage 1: initial opcode-routed experts (all ops)
  for (int c = 0; c <= two; ++c) {
    const int o = c ? op1 : op0;
    const int m = two ? (c ? 2 : 1) : 3;
    ffn_blk(Xs, Hs, Xs, Wi1 + o*8192, bi1 + o*128, Wi2 + o*8192, bi2 + o*64, m);
  }

  // Stages 2-3: 7x (attn + carry experts), only for carry ops
  const int cm = (is_carry(op0) ? 1 : 0) | (is_carry(op1) ? 2 : 0);
  if (cm) {
    for (int it = 0; it < NCASC; ++it) {
      attn_blk(Xs, Qs, Ks, Vs, Ss, Ys, Wq, Wk, Wv);
      for (int c = 0; c <= two; ++c) {
        const int o = c ? op1 : op0;
        const int m = two ? (c ? 2 : 1) : 3;
        if (cm & m)
          ffn_blk(Ys, Hs, Xs, Wc1 + o*8192, bc1 + o*128, Wc2 + o*8192, bc2 + o*64, m & cm);
      }
    }
  }

  // 16 DIV iteration FFNs (distinct weights), DIV rows only
  const int dm = ((op0 == DIV_OP) ? 1 : 0) | ((op1 == DIV_OP) ? 2 : 0);
  if (dm)
    for (int i = 0; i < NITER; ++i)
      ffn_blk(Xs, Hs, Xs, Wd1 + i*8192, bd1 + i*128, Wd2 + i*8192, bd2 + i*64, dm);

  // 16 MOD iteration FFNs, MOD rows only
  const int mm = ((op0 == MOD_OP) ? 1 : 0) | ((op1 == MOD_OP) ? 2 : 0);
  if (mm)
    for (int i = 0; i < NITER; ++i)
      ffn_blk(Xs, Hs, Xs, Wm1 + i*8192, bm1 + i*128, Wm2 + i*8192, bm2 + i*64, mm);

  // Stage 4: finalization experts (all ops)
  for (int c = 0; c <= two; ++c) {
    const int o = c ? op1 : op0;
    const int m = two ? (c ? 2 : 1) : 3;
    ffn_blk(Xs, Hs, Xs, Wf1 + o*8192, bf1 + o*128, Wf2 + o*8192, bf2 + o*64, m);
  }

  // store pair
  {
    float* d0 = out + (size_t)e0 * 512;
    float* d1 = out + (size_t)e1 * 512;
    for (int idx = lane; idx < 256; idx += 32) {
      const int row = idx >> 4;
      const int c4  = (idx & 15) * 4;
      float4 v = *(const float4*)(Xs + row * XPITCH + c4);
      float* dst = (row < 8) ? (d0 + row * 64 + c4) : (d1 + (row - 8) * 64 + c4);
      *(float4*)dst = v;
    }
  }
}

extern "C" void kernel_launch(void* const* d_in, const int* in_sizes, int n_in,
                              void* d_out, int out_size, void* d_ws, size_t ws_size,
                              hipStream_t stream) {
  const float* x   = (const float*)d_in[0];
  const float* Wi1 = (const float*)d_in[1];
  const float* bi1 = (const float*)d_in[2];
  const float* Wi2 = (const float*)d_in[3];
  const float* bi2 = (const float*)d_in[4];
  const float* Wq  = (const float*)d_in[5];
  const float* Wk  = (const float*)d_in[6];
  const float* Wv  = (const float*)d_in[7];
  const float* Wc1 = (const float*)d_in[8];
  const float* bc1 = (const float*)d_in[9];
  const float* Wc2 = (const float*)d_in[10];
  const float* bc2 = (const float*)d_in[11];
  const float* Wd1 = (const float*)d_in[12];
  const float* bd1 = (const float*)d_in[13];
  const float* Wd2 = (const float*)d_in[14];
  const float* bd2 = (const float*)d_in[15];
  const float* Wm1 = (const float*)d_in[16];
  const float* bm1 = (const float*)d_in[17];
  const float* Wm2 = (const float*)d_in[18];
  const float* bm2 = (const float*)d_in[19];
  const float* Wf1 = (const float*)d_in[20];
  const float* bf1 = (const float*)d_in[21];
  const float* Wf2 = (const float*)d_in[22];
  const float* bf2 = (const float*)d_in[23];

  // workspace: op[B] | perm[B] | counts[37] | offs[38] | cursor[37]  (~128.5 KB)
  int* opArr  = (int*)d_ws;
  int* perm   = opArr + B_N;
  int* counts = perm + B_N;
  int* offs   = counts + NOPS;
  int* cursor = offs + NOPS + 1;

  k_zero   <<<1, 64, 0, stream>>>(counts, cursor);
  k_route  <<<B_N / 256, 256, 0, stream>>>(x, opArr, counts);
  k_scan   <<<1, 32, 0, stream>>>(counts, offs);
  k_scatter<<<B_N / 256, 256, 0, stream>>>(opArr, offs, cursor, perm);

  k_main<<<B_N / (2 * WPB), WPB * 32, 0, stream>>>(
      x, Wi1, bi1, Wi2, bi2, Wq, Wk, Wv, Wc1, bc1, Wc2, bc2,
      Wd1, bd1, Wd2, bd2, Wm1, bm1, Wm2, bm2, Wf1, bf1, Wf2, bf2,
      (float*)d_out, opArr, perm);
}